// YOLOXLoss_9216999817659
// MI455X (gfx1250) — compile-verified
//
#include <hip/hip_runtime.h>
#include <hip/hip_bf16.h>

// ---------------- problem constants (from reference) ----------------
#define B_IMG   32
#define N_GT_C  50
#define IMG_C   640
#define NA_C    8400          // 80*80 + 40*40 + 20*20
#define BLK     256
#define BPI     33            // blocks per image: ceil(8400/256)
#define NQ      6             // reduced quantities per anchor

typedef float v2f __attribute__((ext_vector_type(2)));
typedef float v8f __attribute__((ext_vector_type(8)));

// ---------------- WMMA-based wave32 sum (exact f32) ----------------
// A = 16x4 with per-lane (x, 0); B = 4x16 all-ones; D rows = row-sums of A.
// Sum of all D elements seen by a lane (8 regs) + xor-16 shuffle = full
// 32-lane sum of x, independent of the exact lane<->(M,K) mapping.
__device__ __forceinline__ float wave_sum_wmma(float x) {
    v2f a; a[0] = x;    a[1] = 0.0f;
    v2f b; b[0] = 1.0f; b[1] = 1.0f;
    v8f c = {};
    v8f d = __builtin_amdgcn_wmma_f32_16x16x4_f32(
        /*neg_a=*/false, a, /*neg_b=*/false, b,
        /*c_mod=*/(short)0, c, /*reuse_a=*/false, /*reuse_b=*/false);
    float s = d[0] + d[1] + d[2] + d[3] + d[4] + d[5] + d[6] + d[7];
    s += __shfl_xor(s, 16, 32);   // combine the two 8-row halves
    return s;                      // every lane holds the wave total
}

// softplus(l) - l*y, numerically stable (matches jax.nn.softplus BCE)
__device__ __forceinline__ float bce(float l, float y) {
    return fmaxf(l, 0.0f) - l * y + log1pf(expf(-fabsf(l)));
}

// ---------------- main kernel: per-anchor math + block partial sums ----------------
// acc layout: acc[(b*BPI + bx)*NQ + q], q: 0=iou 1=obj 2=cls 3=mask 4=imask 5=fg
__global__ void yolox_main(const float* __restrict__ targets,      // [B,50,5]
                           const float* __restrict__ output,       // [B,NA,7]
                           const unsigned char* __restrict__ fg_mask, // [B,NA] bool
                           const int* __restrict__ matched,        // [B,NA]
                           const float* __restrict__ pred_ious,    // [B,NA]
                           const float* __restrict__ masks,        // [B,640,640,2]
                           float* __restrict__ acc) {
    const int b = blockIdx.y;
    const int a = blockIdx.x * BLK + threadIdx.x;

    float q0 = 0.0f, q1 = 0.0f, q2 = 0.0f, q3 = 0.0f, q4 = 0.0f, q5 = 0.0f;

    if (a < NA_C) {
        // FPN level decode: sizes 80,40,20 -> strides k = 8,16,32
        int off, s;
        if (a < 6400)      { off = 0;    s = 80; }
        else if (a < 8000) { off = 6400; s = 40; }
        else               { off = 8000; s = 20; }
        const int k   = IMG_C / s;              // 8,16,32 (exact)
        const int loc = a - off;
        const int i   = loc / s;
        const int j   = loc - i * s;
        const int row = i * k + (k >> 1);       // nearest-neighbor pixel center
        const int col = j * k + (k >> 1);       // always in [0,639]

        // gather the two mask channels for this anchor (8 contiguous bytes)
        const float* mp = masks + (((size_t)b * IMG_C + row) * IMG_C + col) * 2;
        const float m0 = mp[0];
        const float m1 = mp[1];

        const int idx = b * NA_C + a;
        const float fgf = fg_mask[idx] ? 1.0f : 0.0f;

        const float* o = output + (size_t)idx * 7;
        const float px = o[0], py = o[1], pw = o[2], ph = o[3];
        const float mlog = o[4], olog = o[5], clog = o[6];

        const int g = matched[idx];
        const float* t = targets + ((size_t)b * N_GT_C + g) * 5 + 1;
        const float gx = t[0], gy = t[1], gw = t[2], gh = t[3];

        // IOU loss (YOLOX 'iou' type): 1 - iou^2
        const float tlx = fmaxf(px - pw * 0.5f, gx - gw * 0.5f);
        const float tly = fmaxf(py - ph * 0.5f, gy - gh * 0.5f);
        const float brx = fminf(px + pw * 0.5f, gx + gw * 0.5f);
        const float bry = fminf(py + ph * 0.5f, gy + gh * 0.5f);
        const float area_p = pw * ph;
        const float area_g = gw * gh;
        const float en = (tlx < brx && tly < bry) ? 1.0f : 0.0f;
        const float area_i = (brx - tlx) * (bry - tly) * en;
        const float iou = area_i / (area_p + area_g - area_i + 1e-16f);

        q0 = (1.0f - iou * iou) * fgf;               // iou_per
        q1 = bce(olog, fgf);                          // obj_per
        q2 = bce(clog, pred_ious[idx]) * fgf;         // cls_per
        const float im = (m0 + m1 > 0.0f) ? 1.0f : 0.0f;
        q3 = bce(mlog, m1) * im;                      // mask_per
        q4 = im;                                      // imask
        q5 = fgf;                                     // fg count
    }

    // wave-level reduction via WMMA (EXEC is all-ones here: 8 full waves/block)
    const float s0 = wave_sum_wmma(q0);
    const float s1 = wave_sum_wmma(q1);
    const float s2 = wave_sum_wmma(q2);
    const float s3 = wave_sum_wmma(q3);
    const float s4 = wave_sum_wmma(q4);
    const float s5 = wave_sum_wmma(q5);

    __shared__ float lds[8][NQ];
    const int wave = threadIdx.x >> 5;
    const int lane = threadIdx.x & 31;
    if (lane == 0) {
        lds[wave][0] = s0; lds[wave][1] = s1; lds[wave][2] = s2;
        lds[wave][3] = s3; lds[wave][4] = s4; lds[wave][5] = s5;
    }
    __syncthreads();
    if (threadIdx.x < NQ) {
        float tsum = 0.0f;
        #pragma unroll
        for (int w = 0; w < 8; ++w) tsum += lds[w][threadIdx.x];
        acc[((size_t)(b * BPI + blockIdx.x)) * NQ + threadIdx.x] = tsum;
    }
}

// ---------------- finalize: deterministic tree over block partials ----------------
__global__ void yolox_final(const float* __restrict__ targets,
                            const float* __restrict__ acc,
                            float* __restrict__ out) {
    __shared__ float sb[B_IMG][NQ];
    __shared__ float ngt[B_IMG];
    const int t = threadIdx.x;

    if (t < B_IMG) {
        // per-image reduction over the 33 block partials (fixed order)
        float s[NQ] = {0, 0, 0, 0, 0, 0};
        for (int x = 0; x < BPI; ++x) {
            const float* p = acc + ((size_t)(t * BPI + x)) * NQ;
            #pragma unroll
            for (int q = 0; q < NQ; ++q) s[q] += p[q];
        }
        #pragma unroll
        for (int q = 0; q < NQ; ++q) sb[t][q] = s[q];

        int cnt = 0;
        for (int g = 0; g < N_GT_C; ++g)
            cnt += (targets[((size_t)t * N_GT_C + g) * 5] > 0.0f) ? 1 : 0;
        ngt[t] = (float)cnt;
    }
    __syncthreads();

    if (t == 0) {
        float iou_sum = 0, obj_sum = 0, cls_sum = 0, msk_sum = 0, im_sum = 0,
              fg_sum = 0, gt_sum = 0;
        for (int b = 0; b < B_IMG; ++b) {
            iou_sum += sb[b][0];
            obj_sum += sb[b][1];
            cls_sum += sb[b][2];
            msk_sum += sb[b][3];
            im_sum  += sb[b][4];
            fg_sum  += sb[b][5];
            gt_sum  += ngt[b];
        }
        const float num_f = fmaxf(fg_sum, 1.0f);
        const float num_g = fmaxf(gt_sum, 1.0f);
        const float num_m = fmaxf(im_sum, 1.0f);
        const float iou_loss  = 5.0f * iou_sum / num_f;
        const float obj_loss  = obj_sum / num_f;
        const float cls_loss  = cls_sum / num_f;
        const float mask_loss = 2.0f * msk_sum / num_m;
        out[0] = iou_loss + obj_loss + cls_loss + mask_loss;
        out[1] = iou_loss;
        out[2] = obj_loss;
        out[3] = cls_loss;
        out[4] = mask_loss;
        out[37] = num_f / num_g;
    }
    if (t < B_IMG) {
        const float ob = sb[t][1], cl = sb[t][2], fg = sb[t][5];
        out[5 + t] = (ngt[t] > 0.0f) ? (ob + cl) / fmaxf(fg, 1.0f) : -1.0f;
    }
}

extern "C" void kernel_launch(void* const* d_in, const int* in_sizes, int n_in,
                              void* d_out, int out_size, void* d_ws, size_t ws_size,
                              hipStream_t stream) {
    const float*         targets   = (const float*)d_in[0];        // [32,50,5] f32
    const float*         output    = (const float*)d_in[1];        // [32,8400,7] f32
    const unsigned char* fg_mask   = (const unsigned char*)d_in[2];// [32,8400] bool
    const int*           matched   = (const int*)d_in[3];          // [32,8400] i32
    const float*         pred_ious = (const float*)d_in[4];        // [32,8400] f32
    const float*         masks     = (const float*)d_in[5];        // [32,640,640,2] f32
    float* acc = (float*)d_ws;     // 32*33*6 floats of block partials
    float* out = (float*)d_out;    // 38 floats

    dim3 grid(BPI, B_IMG);
    yolox_main<<<grid, BLK, 0, stream>>>(targets, output, fg_mask, matched,
                                         pred_ious, masks, acc);
    yolox_final<<<1, 64, 0, stream>>>(targets, acc, out);
}